// PreferenceRouter_46368466927849
// MI455X (gfx1250) — compile-verified
//
#include <hip/hip_runtime.h>
#include <stdint.h>

typedef __attribute__((ext_vector_type(2))) float v2f;
typedef __attribute__((ext_vector_type(8))) float v8f;

#define HIDDEN   4096
#define SEQ      4096
#define BATCH    4
#define RANK     16
#define NUM_ATTR 8
#define SCALING  (1.0f / 16.0f)

// Phase-1 K staging geometry
#define CHUNK    512                   // K columns per chunk
#define NCHUNK   (HIDDEN / CHUNK)      // 8 chunks
#define LDPAD    4                     // pad -> ds bank stride 4, conflict-free
#define LROW     (CHUNK + LDPAD)       // 516 floats per staged row
#define BUFSZ    (16 * LROW)           // floats per staging buffer (8256)

// ---------------------------------------------------------------------------
// ws layout (floats): [0, 1024) = Wb[b][r][i]
//                     [1024, 1024 + 4*4096*16) = Beff[b][d][i] = B_w x Wb[b]
// ---------------------------------------------------------------------------

__global__ void wb_kernel(const float* __restrict__ alphas,
                          const float* __restrict__ W,
                          float* __restrict__ wb) {
    const int b = blockIdx.x;
    const int t = threadIdx.x;          // t = r*16 + i
    float s = 0.f;
#pragma unroll
    for (int k = 0; k < NUM_ATTR; ++k)
        s += alphas[b * NUM_ATTR + k] * W[k * RANK * RANK + t];
    wb[b * RANK * RANK + t] = s;
}

__global__ void beff_kernel(const float* __restrict__ Bw,
                            const float* __restrict__ wb,
                            float* __restrict__ beff) {
    const int g = blockIdx.x * blockDim.x + threadIdx.x;  // [0, 4*4096*16)
    const int i = g & 15;
    const int d = (g >> 4) & (HIDDEN - 1);
    const int b = g >> 16;
    float s = 0.f;
#pragma unroll
    for (int r = 0; r < RANK; ++r)
        s += Bw[d * RANK + r] * wb[b * RANK * RANK + r * RANK + i];
    beff[g] = s;
}

// ---------------------------------------------------------------------------
// CDNA5 helpers: async global->LDS copy (ASYNCcnt) + waits
// ---------------------------------------------------------------------------
__device__ __forceinline__ uint32_t lds_u32(const void* p) {
    return (uint32_t)(uintptr_t)(__attribute__((address_space(3))) const char*)p;
}

__device__ __forceinline__ void async_ld_b128(uint32_t lds_dst, uint32_t voff,
                                              const float* sbase) {
    // GVS mode: mem = SGPR64 + VGPR_I32 ; dest = per-lane LDS byte address
    asm volatile("global_load_async_to_lds_b128 %0, %1, %2"
                 :: "v"(lds_dst), "v"(voff), "s"(sbase)
                 : "memory");
}
__device__ __forceinline__ void wait_async_le16() {
    asm volatile("s_wait_asynccnt 0x10" ::: "memory");
}
__device__ __forceinline__ void wait_async_0() {
    asm volatile("s_wait_asynccnt 0x0" ::: "memory");
}

// ---------------------------------------------------------------------------
// Main fused kernel: one block = 16 consecutive rows of h.
// Phase 1: P = h_tile * A_w^T   (async double-buffered LDS staging + WMMA f32)
// Phase 2: cross-wave reduce, scale by 1/16, store P row-major in LDS
// Phase 3: out_tile = h_tile + P * Beff[b]^T
// ---------------------------------------------------------------------------
__global__ __launch_bounds__(256)
void router_kernel(const float* __restrict__ h,
                   const float* __restrict__ Aw,
                   const float* __restrict__ beff,
                   float* __restrict__ out) {
    // slots 0/1: h chunk double buffer; slots 2/3: A_w chunk double buffer
    __shared__ float lds[4 * BUFSZ];               // 132,096 bytes
    float* lds_red = lds;                          // aliased after staging done
    float* lds_P   = lds + 2048;

    const int t      = blockIdx.x;                 // 1024 tiles
    const int b      = t >> 8;
    const int s0     = (t & 255) << 4;
    const int tid    = threadIdx.x;
    const int wave   = tid >> 5;
    const int lane   = tid & 31;
    const int lane16 = lane & 15;
    const int half   = lane >> 4;

    const size_t rowbase = (size_t)b * SEQ + (size_t)s0;
    const float* sb_h = h + rowbase * HIDDEN;      // 16 x 4096 tile base
    const float* sb_a = Aw;                        // 16 x 4096, same geometry
    const uint32_t lbase = lds_u32(lds);

    // Issue one chunk's async loads (both matrices) into buffer slot s.
    // 16 rows x 512 floats = 2048 b128 segments per matrix -> 8 per thread.
    auto issue = [&](int c, int s) {
#pragma unroll
        for (int i = 0; i < 8; ++i) {
            const int seg = i * 256 + tid;
            const int row = seg >> 7;              // 128 segs per row
            const int c4  = (seg & 127) << 2;      // float col within chunk
            const uint32_t voff =
                (uint32_t)((row * HIDDEN + c * CHUNK + c4) * 4);
            const uint32_t dst =
                lbase + (uint32_t)((s * BUFSZ + row * LROW + c4) * 4);
            async_ld_b128(dst, voff, sb_h);
            async_ld_b128(dst + 2u * BUFSZ * 4u, voff, sb_a);
        }
    };

    // ---- Phase 1: software-pipelined staging + WMMA -----------------------
    issue(0, 0);
    issue(1, 1);
    wait_async_le16();                             // chunk 0 landed (per wave)
    __syncthreads();                               // landed for all waves

    v8f acc = {};
    for (int c = 0; c < NCHUNK; ++c) {
        const int s = c & 1;
        // wave w owns columns [w*64, w*64+64) of each chunk
        const float* hb = lds + s * BUFSZ + lane16 * LROW + wave * 64 + 2 * half;
        const float* ab = hb + 2 * BUFSZ;
#pragma unroll
        for (int ks = 0; ks < 64; ks += 4) {
            v2f a  = *(const v2f*)(hb + ks);       // ds_load_b64, conflict-free
            v2f bw = *(const v2f*)(ab + ks);
            acc = __builtin_amdgcn_wmma_f32_16x16x4_f32(
                false, a, false, bw, (short)0, acc, false, false);
        }
        if (c < NCHUNK - 1) {
            __syncthreads();                       // all waves done with slot s
            if (c + 2 < NCHUNK) { issue(c + 2, s); wait_async_le16(); }
            else                { wait_async_0(); }
            __syncthreads();                       // chunk c+1 visible to all
        }
    }
    __syncthreads();                               // buffers free; allow alias

    // ---- store per-wave partials (C-layout packed) ------------------------
#pragma unroll
    for (int j = 0; j < 8; ++j)
        lds_red[wave * 256 + lane * 8 + j] = acc[j];
    __syncthreads();

    // ---- Phase 2: reduce across waves; one P element per thread -----------
    {
        const int m   = tid >> 4;
        const int n   = tid & 15;
        const int idx = (n + ((m >> 3) << 4)) * 8 + (m & 7);
        float s = 0.f;
#pragma unroll
        for (int w = 0; w < 8; ++w) s += lds_red[w * 256 + idx];
        lds_P[m * 16 + n] = s * SCALING;
    }
    __syncthreads();

    // ---- Phase 3: out = h + P_scaled * Beff_b^T ---------------------------
    v2f afrag[4];
#pragma unroll
    for (int kk = 0; kk < 4; ++kk)
        afrag[kk] = *(const v2f*)(lds_P + lane16 * 16 + kk * 4 + 2 * half);

    const float* beff_b = beff + (size_t)b * HIDDEN * RANK;

    for (int nt = wave; nt < 256; nt += 8) {
        const int n0 = nt << 4;

        const float* hbp = h + rowbase * HIDDEN + n0;
        v8f cacc;
#pragma unroll
        for (int j = 0; j < 8; ++j)
            cacc[j] = hbp[(size_t)(j + 8 * half) * HIDDEN + lane16];

        const float* bb = beff_b + (size_t)(n0 + lane16) * RANK + 2 * half;
#pragma unroll
        for (int kk = 0; kk < 4; ++kk) {
            v2f bf = *(const v2f*)(bb + kk * 4);
            cacc = __builtin_amdgcn_wmma_f32_16x16x4_f32(
                false, afrag[kk], false, bf, (short)0, cacc, false, false);
        }

        float* ob = out + rowbase * HIDDEN + n0;
#pragma unroll
        for (int j = 0; j < 8; ++j)
            ob[(size_t)(j + 8 * half) * HIDDEN + lane16] = cacc[j];
    }
}

// ---------------------------------------------------------------------------
extern "C" void kernel_launch(void* const* d_in, const int* in_sizes, int n_in,
                              void* d_out, int out_size, void* d_ws, size_t ws_size,
                              hipStream_t stream) {
    const float* h      = (const float*)d_in[0];
    const float* alphas = (const float*)d_in[1];
    const float* Aw     = (const float*)d_in[2];
    const float* Bw     = (const float*)d_in[3];
    const float* W      = (const float*)d_in[4];
    float*       out    = (float*)d_out;

    float* wb   = (float*)d_ws;                    // 4*16*16 floats
    float* beff = wb + BATCH * RANK * RANK;        // 4*4096*16 floats

    wb_kernel<<<BATCH, RANK * RANK, 0, stream>>>(alphas, W, wb);
    beff_kernel<<<(BATCH * HIDDEN * RANK) / 256, 256, 0, stream>>>(Bw, wb, beff);
    router_kernel<<<(BATCH * SEQ) / 16, 256, 0, stream>>>(h, Aw, beff, out);
}